// VSC_53352083751229
// MI455X (gfx1250) — compile-verified
//
#include <hip/hip_runtime.h>

// Problem constants (fixed by the reference)
#define BATCH   64
#define LTOK    2048
#define DDIM    384
#define KKEEP   1024
#define NCHUNK  8          // L-chunks for the mean reduction
#define CHUNKL  (LTOK / NCHUNK)

typedef float v2f __attribute__((ext_vector_type(2)));
typedef float v8f __attribute__((ext_vector_type(8)));

// ---------------------------------------------------------------------------
// Kernel 1: partial sums over L for the mean  (reads the whole 201MB tensor)
// grid = (NCHUNK, BATCH), block = DDIM threads. Coalesced across d.
// ---------------------------------------------------------------------------
__global__ __launch_bounds__(DDIM) void vsc_mean_partial(
    const float* __restrict__ x, float* __restrict__ partial) {
  const int c = blockIdx.x;
  const int b = blockIdx.y;
  const int d = threadIdx.x;
  const float* p = x + ((size_t)b * LTOK + (size_t)c * CHUNKL) * DDIM + d;
  float acc = 0.0f;
  #pragma unroll 4
  for (int l = 0; l < CHUNKL; ++l) acc += p[(size_t)l * DDIM];
  partial[((size_t)b * NCHUNK + c) * DDIM + d] = acc;
}

// ---------------------------------------------------------------------------
// Kernel 2: m[b,d] = mean + t_cls
// ---------------------------------------------------------------------------
__global__ __launch_bounds__(DDIM) void vsc_mean_finish(
    const float* __restrict__ partial, const float* __restrict__ t_cls,
    float* __restrict__ m) {
  const int b = blockIdx.x;
  const int d = threadIdx.x;
  float acc = 0.0f;
  #pragma unroll
  for (int c = 0; c < NCHUNK; ++c)
    acc += partial[((size_t)b * NCHUNK + c) * DDIM + d];
  m[b * DDIM + d] = acc * (1.0f / (float)LTOK) + t_cls[b * DDIM + d];
}

// ---------------------------------------------------------------------------
// Kernel 3: scores via fp32 WMMA (V_WMMA_F32_16X16X4_F32).
// One wave (32 lanes) per 16-token tile; EXEC is all-1s for the WMMA body.
// A (16x4 fp32): lanes 0-15 carry K=0,1 of token M=lane; lanes 16-31 carry
// K=2,3 of token M=lane-16 (ISA 7.12.2 layout). B (4x16) is the m-chunk
// broadcast across all 16 columns, so every column of D equals the dot.
// Per-lane squared-norm partials are merged with shfl_xor(16).
// ---------------------------------------------------------------------------
__global__ __launch_bounds__(32) void vsc_score_wmma(
    const float* __restrict__ x, const float* __restrict__ m,
    float* __restrict__ scores) {
  const int tile = blockIdx.x;          // 0 .. LTOK/16 - 1
  const int b    = blockIdx.y;
  const int lane = threadIdx.x;         // 0..31, wave32

  __shared__ float ms[DDIM];
  for (int i = lane; i < DDIM; i += 32) ms[i] = m[b * DDIM + i];
  __syncthreads();

  const int l0   = tile * 16;
  const int tok  = l0 + (lane & 15);
  const int koff = (lane < 16) ? 0 : 2;
  const float* row = x + ((size_t)b * LTOK + tok) * DDIM;

  v8f c;
  #pragma unroll
  for (int i = 0; i < 8; ++i) c[i] = 0.0f;
  float sq = 0.0f;

  #pragma unroll 4
  for (int k0 = 0; k0 < DDIM; k0 += 4) {
    v2f a, bv;
    a.x  = row[k0 + koff];
    a.y  = row[k0 + koff + 1];
    bv.x = ms[k0 + koff];
    bv.y = ms[k0 + koff + 1];
    sq = fmaf(a.x, a.x, sq);
    sq = fmaf(a.y, a.y, sq);
    // D = A(16x4) * B(4x16) + C   (fp32 throughout)
    c = __builtin_amdgcn_wmma_f32_16x16x4_f32(
        /*neg_a=*/false, a, /*neg_b=*/false, bv,
        /*c_mod=*/(short)0, c, /*reuse_a=*/false, /*reuse_b=*/false);
  }

  // token M (M<16) full squared norm lands in lane M (dup in lane M+16)
  sq += __shfl_xor(sq, 16, 32);

  // C/D layout: lane 0 holds dots of tokens 0..7 in c[0..7];
  //             lane 16 holds dots of tokens 8..15 in c[0..7].
  const int base = (lane < 16) ? 0 : 8;
  #pragma unroll
  for (int t = 0; t < 8; ++t) {
    float n2 = __shfl(sq, base + t, 32);   // uniform shuffle, all lanes join
    if (lane == 0 || lane == 16) {
      float nrm = fmaxf(sqrtf(n2), 1e-12f);  // matches F.normalize eps
      scores[(size_t)b * LTOK + l0 + base + t] = c[t] / nrm;
    }
  }
}

// ---------------------------------------------------------------------------
// Kernel 4: per-batch bitonic sort of (score, index) pairs in LDS,
// descending score, ties broken by ascending index (== lax.top_k order).
// 2048 elems -> 16KB LDS, 1024 threads, 66 compare stages.
// ---------------------------------------------------------------------------
__global__ __launch_bounds__(1024) void vsc_topk_sort(
    const float* __restrict__ scores, int* __restrict__ idx) {
  const int b   = blockIdx.x;
  const int tid = threadIdx.x;
  __shared__ float s[LTOK];
  __shared__ int   id[LTOK];
  for (int i = tid; i < LTOK; i += 1024) {
    s[i]  = scores[(size_t)b * LTOK + i];
    id[i] = i;
  }
  __syncthreads();
  for (int k = 2; k <= LTOK; k <<= 1) {
    for (int j = k >> 1; j > 0; j >>= 1) {
      for (int i = tid; i < LTOK; i += 1024) {
        const int ixj = i ^ j;
        if (ixj > i) {
          const float si = s[i], sj = s[ixj];
          const int   ii = id[i], ij = id[ixj];
          const bool jBetter = (sj > si) || (sj == si && ij < ii);
          const bool iBetter = (si > sj) || (si == sj && ii < ij);
          const bool doSwap  = ((i & k) == 0) ? jBetter : iBetter;
          if (doSwap) { s[i] = sj; s[ixj] = si; id[i] = ij; id[ixj] = ii; }
        }
      }
      __syncthreads();
    }
  }
  for (int i = tid; i < KKEEP; i += 1024) idx[b * KKEEP + i] = id[i];
}

// ---------------------------------------------------------------------------
// Kernel 5: gather kept rows, float4 (b128) transfers.
// grid = B*K blocks of 96 threads (one float4 per thread per 384-float row).
// ---------------------------------------------------------------------------
__global__ __launch_bounds__(96) void vsc_gather_rows(
    const float4* __restrict__ x4, const int* __restrict__ idx,
    float4* __restrict__ out4) {
  const int r = blockIdx.x;            // 0 .. B*K-1
  const int b = r >> 10;               // / KKEEP
  const int src = idx[r];
  const int t = threadIdx.x;           // 0..95
  out4[(size_t)r * (DDIM / 4) + t] =
      x4[((size_t)b * LTOK + src) * (DDIM / 4) + t];
}

// ---------------------------------------------------------------------------
extern "C" void kernel_launch(void* const* d_in, const int* in_sizes, int n_in,
                              void* d_out, int out_size, void* d_ws, size_t ws_size,
                              hipStream_t stream) {
  const float* x     = (const float*)d_in[0];   // [64, 2048, 384] fp32
  const float* t_cls = (const float*)d_in[1];   // [64, 384] fp32
  float*       out   = (float*)d_out;           // [64, 1024, 384] fp32

  char* ws = (char*)d_ws;
  const size_t partial_bytes = (size_t)BATCH * NCHUNK * DDIM * sizeof(float); // 768KB
  const size_t m_bytes       = (size_t)BATCH * DDIM * sizeof(float);          // 96KB
  const size_t score_bytes   = (size_t)BATCH * LTOK * sizeof(float);          // 512KB
  float* partial = (float*)ws;
  float* m       = (float*)(ws + partial_bytes);
  float* scores  = (float*)(ws + partial_bytes + m_bytes);
  int*   idx     = (int*)  (ws + partial_bytes + m_bytes + score_bytes);

  vsc_mean_partial<<<dim3(NCHUNK, BATCH), DDIM, 0, stream>>>(x, partial);
  vsc_mean_finish <<<BATCH, DDIM, 0, stream>>>(partial, t_cls, m);
  vsc_score_wmma  <<<dim3(LTOK / 16, BATCH), 32, 0, stream>>>(x, m, scores);
  vsc_topk_sort   <<<BATCH, 1024, 0, stream>>>(scores, idx);
  vsc_gather_rows <<<BATCH * KKEEP, 96, 0, stream>>>((const float4*)x, idx,
                                                     (float4*)out);
}